// SphericalHarmonics_39161511805246
// MI455X (gfx1250) — compile-verified
//
#include <hip/hip_runtime.h>
#include <stdint.h>

#define TPB 256
#define PTS_PER_BLOCK 256

typedef float v4f __attribute__((ext_vector_type(4)));

__device__ __forceinline__ void sh_compute(float vx, float vy, float vz,
                                           float* sh1, float* sh2, float* sh3) {
    float sn  = vx * vx + vy * vy + vz * vz;
    float inv = (sn == 0.0f) ? 0.0f : rsqrtf(sn);
    float ux = vx * inv, uy = vy * inv, uz = vz * inv;
    float x2 = ux * ux, y2 = uy * uy, z2 = uz * uz;
    float x2z2 = x2 + z2;

    const float S3     = 1.7320508075688772f;   // sqrt(3)
    const float S5     = 2.2360679774997896f;   // sqrt(5)
    const float S7     = 2.6457513110645906f;   // sqrt(7)
    const float S15    = 3.8729833462074170f;   // sqrt(15)
    const float C42_6  = 1.0801234497346435f;   // sqrt(42)/6
    const float C168_8 = 1.6201851746019651f;   // sqrt(168)/8

    sh1[0] = S3 * ux;
    sh1[1] = S3 * uy;
    sh1[2] = S3 * uz;

    float sh2_0 = S15 * ux * uz;
    float sh2_4 = 0.5f * S15 * (z2 - x2);
    sh2[0] = sh2_0;
    sh2[1] = S15 * ux * uy;
    sh2[2] = S5 * (y2 - 0.5f * x2z2);
    sh2[3] = S15 * uy * uz;
    sh2[4] = sh2_4;

    float t = 4.0f * y2 - x2z2;
    sh3[0] = C42_6 * (sh2_0 * uz + sh2_4 * ux);
    sh3[1] = S7 * sh2_0 * uy;
    sh3[2] = C168_8 * t * ux;
    sh3[3] = 0.5f * S7 * uy * (2.0f * y2 - 3.0f * x2z2);
    sh3[4] = C168_8 * uz * t;
    sh3[5] = S7 * sh2_4 * uy;
    sh3[6] = C42_6 * (sh2_4 * uz - sh2_0 * ux);
}

// LDS layout (floats):
//   s_in : [   0,  768)   256 pts * 3
//   s_o1 : [ 768, 1536)   256 pts * 3
//   s_o2 : [1536, 2816)   256 pts * 5
//   s_o3 : [2816, 4608)   256 pts * 7
// All region starts are 16B aligned.
__global__ __launch_bounds__(TPB) void SphericalHarmonics_kernel(
    const float* __restrict__ x, float* __restrict__ out, int n) {
    __shared__ __align__(16) float smem[4608];
    float* s_in = smem;
    float* s_o1 = smem + 768;
    float* s_o2 = smem + 1536;
    float* s_o3 = smem + 2816;

    const int tid = threadIdx.x;
    const long long base = (long long)blockIdx.x * PTS_PER_BLOCK;

    float* o0 = out;                       // N   floats, sh0
    float* o1 = out + (long long)n;        // 3N  floats, sh1
    float* o2 = out + 4LL * n;             // 5N  floats, sh2
    float* o3 = out + 9LL * n;             // 7N  floats, sh3

    if (base + PTS_PER_BLOCK <= (long long)n) {
        // ---- stage 768 input floats into LDS via CDNA5 async b128 loads ----
        if (tid < 192) {
            unsigned lds_addr = (unsigned)(uintptr_t)(s_in + 4 * tid);
            unsigned long long gaddr =
                (unsigned long long)(uintptr_t)(x + 3 * base) + 16ull * (unsigned)tid;
            asm volatile("global_load_async_to_lds_b128 %0, %1, off"
                         :: "v"(lds_addr), "v"(gaddr)
                         : "memory");
        }
        asm volatile("s_wait_asynccnt 0" ::: "memory");
        __syncthreads();  // all waves' async writes visible to whole block

        // ---- compute (stride-3/5/7 LDS accesses: gcd with 64 banks == 1) ----
        float vx = s_in[3 * tid + 0];
        float vy = s_in[3 * tid + 1];
        float vz = s_in[3 * tid + 2];

        float sh1[3], sh2[5], sh3[7];
        sh_compute(vx, vy, vz, sh1, sh2, sh3);

#pragma unroll
        for (int k = 0; k < 3; ++k) s_o1[3 * tid + k] = sh1[k];
#pragma unroll
        for (int k = 0; k < 5; ++k) s_o2[5 * tid + k] = sh2[k];
#pragma unroll
        for (int k = 0; k < 7; ++k) s_o3[7 * tid + k] = sh3[k];
        __syncthreads();

        // ---- fully coalesced b128 non-temporal stores (1024 v4f/block) ----
        v4f* p0 = reinterpret_cast<v4f*>(o0 + base);
        v4f* p1 = reinterpret_cast<v4f*>(o1 + 3 * base);
        v4f* p2 = reinterpret_cast<v4f*>(o2 + 5 * base);
        v4f* p3 = reinterpret_cast<v4f*>(o3 + 7 * base);
        const v4f* q1 = reinterpret_cast<const v4f*>(s_o1);
        const v4f* q2 = reinterpret_cast<const v4f*>(s_o2);
        const v4f* q3 = reinterpret_cast<const v4f*>(s_o3);

        const v4f ones = {1.0f, 1.0f, 1.0f, 1.0f};
        for (int j = tid; j < 64; j += TPB)  __builtin_nontemporal_store(ones,  p0 + j);
        for (int j = tid; j < 192; j += TPB) __builtin_nontemporal_store(q1[j], p1 + j);
        for (int j = tid; j < 320; j += TPB) __builtin_nontemporal_store(q2[j], p2 + j);
        for (int j = tid; j < 448; j += TPB) __builtin_nontemporal_store(q3[j], p3 + j);
    } else {
        // ---- tail: scalar per-point path ----
        long long p = base + tid;
        if (p < (long long)n) {
            float vx = x[3 * p + 0], vy = x[3 * p + 1], vz = x[3 * p + 2];
            float sh1[3], sh2[5], sh3[7];
            sh_compute(vx, vy, vz, sh1, sh2, sh3);
            o0[p] = 1.0f;
#pragma unroll
            for (int k = 0; k < 3; ++k) o1[3 * p + k] = sh1[k];
#pragma unroll
            for (int k = 0; k < 5; ++k) o2[5 * p + k] = sh2[k];
#pragma unroll
            for (int k = 0; k < 7; ++k) o3[7 * p + k] = sh3[k];
        }
    }
}

extern "C" void kernel_launch(void* const* d_in, const int* in_sizes, int n_in,
                              void* d_out, int out_size, void* d_ws, size_t ws_size,
                              hipStream_t stream) {
    const float* x = (const float*)d_in[0];
    float* out = (float*)d_out;
    const int n = in_sizes[0] / 3;          // N points, 3 floats each
    const int blocks = (n + PTS_PER_BLOCK - 1) / PTS_PER_BLOCK;
    SphericalHarmonics_kernel<<<blocks, TPB, 0, stream>>>(x, out, n);
}